// PQCQuanv_51101520888104
// MI455X (gfx1250) — compile-verified
//
#include <hip/hip_runtime.h>
#include <hip/hip_bf16.h>

typedef __attribute__((ext_vector_type(2))) float v2f;
typedef __attribute__((ext_vector_type(8))) float v8f;

#define N_QUBITS 6
#define D_DIM 64
#define L_LAYERS 10
#define C_OUT 4
#define OH 30
#define OW 30
#define BATCH 32
// total patch-circuit pairs = 32*30*30*4 = 115200 = 14400 blocks * 8 waves

__device__ __forceinline__ float lane_bpermute(float v, int srcLane) {
    // Dst[lane] = src[srcLane]; wave32 uses index bits [6:2]
    return __int_as_float(__builtin_amdgcn_ds_bpermute(srcLane << 2, __float_as_int(v)));
}

__global__ __launch_bounds__(256) void pqc_quanv_kernel(
    const float* __restrict__ x,        // [32,1,32,32]
    const int*   __restrict__ perms,    // [4,10,64]; perm[i] = i ^ m, m = perms[j,l,0]
    const float* __restrict__ phase_re, // [4,10,64]
    const float* __restrict__ phase_im, // [4,10,64]
    const float* __restrict__ bw,       // [4,10]
    const int*   __restrict__ feats,    // [4,10]
    const float* __restrict__ zsign,    // [64]
    float*       __restrict__ out)      // [32,4,30,30]
{
    const int lane = threadIdx.x & 31;
    const int wave = blockIdx.x * 8 + (threadIdx.x >> 5); // 0..115199
    const int j = wave & 3;          // circuit
    const int n = wave >> 2;         // patch id 0..28799
    const int b = n / (OH * OW);
    const int rem = n - b * (OH * OW);
    const int oy = rem / OW;
    const int ox = rem - oy * OW;

    const float* xb = x + b * 1024 + oy * 32 + ox; // top-left pixel of this patch

    // state: amplitude i = lane + 32*r, r in {0,1}; psi0 = e_0
    float re0 = (lane == 0) ? 1.0f : 0.0f, im0 = 0.0f;
    float re1 = 0.0f, im1 = 0.0f;

    const int base = j * L_LAYERS;
    #pragma unroll
    for (int l = 0; l < L_LAYERS; ++l) {
        const int idx = base + l;
        // theta = bandwidth[j,l] * patch_pixel[feats[j,l]]  (wave-uniform)
        const int f  = feats[idx];
        const int ky = f / 3;
        const int kx = f - ky * 3;
        const float th = bw[idx] * xb[ky * 32 + kx];
        const float c = __cosf(0.5f * th);
        const float s = __sinf(0.5f * th);

        // XOR mask of this Pauli layer: perm[0] = 0 ^ m
        const int m   = perms[idx * D_DIM];
        const int m31 = m & 31;
        const bool hi = (m & 32) != 0;

        // phase at destination indices i0 = lane, i1 = lane + 32
        const float pr0 = phase_re[idx * D_DIM + lane];
        const float pi0 = phase_im[idx * D_DIM + lane];
        const float pr1 = phase_re[idx * D_DIM + 32 + lane];
        const float pi1 = phase_im[idx * D_DIM + 32 + lane];

        // gather psi[i ^ m]: register swap for bit5, lane-XOR shuffle for bits[4:0]
        const int sl = (lane ^ m31) << 2;
        const float s0r = hi ? re1 : re0, s0i = hi ? im1 : im0;
        const float s1r = hi ? re0 : re1, s1i = hi ? im0 : im1;
        const float g0r = __int_as_float(__builtin_amdgcn_ds_bpermute(sl, __float_as_int(s0r)));
        const float g0i = __int_as_float(__builtin_amdgcn_ds_bpermute(sl, __float_as_int(s0i)));
        const float g1r = __int_as_float(__builtin_amdgcn_ds_bpermute(sl, __float_as_int(s1r)));
        const float g1i = __int_as_float(__builtin_amdgcn_ds_bpermute(sl, __float_as_int(s1i)));

        // w = -i * s * phase = (s*pi, -s*pr); psi_new = c*psi + w*g
        const float w0r = s * pi0, w0i = -s * pr0;
        const float w1r = s * pi1, w1i = -s * pr1;
        const float nr0 = c * re0 + w0r * g0r - w0i * g0i;
        const float ni0 = c * im0 + w0r * g0i + w0i * g0r;
        const float nr1 = c * re1 + w1r * g1r - w1i * g1i;
        const float ni1 = c * im1 + w1r * g1i + w1i * g1r;
        re0 = nr0; im0 = ni0; re1 = nr1; im1 = ni1;
    }

    // per-lane readout contribution
    const float t = zsign[lane]      * (re0 * re0 + im0 * im0)
                  + zsign[lane + 32] * (re1 * re1 + im1 * im1);

    // Exact 32-lane sum via one V_WMMA_F32_16X16X4_F32:
    // A vgpr0 = t, vgpr1 = 0  =>  A[m,0] = t[m], A[m,2] = t[m+16], others 0
    // B = all ones             =>  D[m,n] = t[m] + t[m+16]
    v2f a;  a[0] = t;    a[1] = 0.0f;
    v2f bo; bo[0] = 1.0f; bo[1] = 1.0f;
    v8f acc = {0.0f, 0.0f, 0.0f, 0.0f, 0.0f, 0.0f, 0.0f, 0.0f};
    acc = __builtin_amdgcn_wmma_f32_16x16x4_f32(false, a, false, bo,
                                                (short)0, acc, false, false);
    // lanes<16 hold D rows 0..7, lanes>=16 hold rows 8..15
    float partial = acc[0] + acc[1] + acc[2] + acc[3]
                  + acc[4] + acc[5] + acc[6] + acc[7];
    const float total = partial + lane_bpermute(partial, lane ^ 16);

    if (lane == 0) {
        out[b * (C_OUT * OH * OW) + j * (OH * OW) + oy * OW + ox] = total;
    }
}

extern "C" void kernel_launch(void* const* d_in, const int* in_sizes, int n_in,
                              void* d_out, int out_size, void* d_ws, size_t ws_size,
                              hipStream_t stream) {
    const float* x        = (const float*)d_in[0];
    const int*   perms    = (const int*)  d_in[1];
    const float* phase_re = (const float*)d_in[2];
    const float* phase_im = (const float*)d_in[3];
    const float* bw       = (const float*)d_in[4];
    const int*   feats    = (const int*)  d_in[5];
    const float* zsign    = (const float*)d_in[6];
    float* out = (float*)d_out;

    // 115200 patch-circuit waves, 8 waves (256 threads) per block
    dim3 grid(14400), block(256);
    pqc_quanv_kernel<<<grid, block, 0, stream>>>(x, perms, phase_re, phase_im,
                                                 bw, feats, zsign, out);
}